// IndexerRef_48825188221289
// MI455X (gfx1250) — compile-verified
//
#include <hip/hip_runtime.h>
#include <hip/hip_bf16.h>

typedef __attribute__((ext_vector_type(2))) float v2f;
typedef __attribute__((ext_vector_type(8))) float v8f;
typedef __attribute__((address_space(3))) float lds_f;

#define B_      64
#define N_      2
#define H_      64
#define D_      128
#define MAXLEN_ 8192
#define BLK_    128
#define NBLK_   64
#define MROWS   128        // N_*H_ query rows per batch
#define LDSPAD  132        // LDS row stride in floats (conflict-free for WMMA fetch)
#define KSEG    8          // k segments (gridDim.x)
#define CHUNKS  8          // 128-wide k chunks per segment (8*8*128 = 8192)

#if __has_builtin(__builtin_amdgcn_s_wait_asynccnt)
#define WAIT_ASYNC(n) __builtin_amdgcn_s_wait_asynccnt(n)
#else
#define WAIT_ASYNC(n) asm volatile("s_wait_asynccnt %0" ::"n"(n) : "memory")
#endif

// Async copy of 16B per lane: global (64-bit VGPR address) -> LDS (32-bit VGPR offset).
// Tracked by ASYNCcnt; data lands in LDS without passing through VGPRs.
__device__ __forceinline__ void async_g2l_b128(unsigned lds_off, const void* gaddr) {
    asm volatile("global_load_async_to_lds_b128 %0, %1, off"
                 :: "v"(lds_off), "v"(gaddr)
                 : "memory");
}

__global__ __launch_bounds__(256)
void indexer_wmma_kernel(const float* __restrict__ q,
                         const float* __restrict__ kv_cache,
                         const float* __restrict__ weights,
                         const int*   __restrict__ context_lens,
                         const int*   __restrict__ block_tables,
                         float* __restrict__ out)
{
    extern __shared__ float smem[];
    float* Qs  = smem;                       // 128 x 132 floats
    float* KV0 = smem + MROWS * LDSPAD;      // 128 x 132 floats (ping)
    float* KV1 = smem + 2 * MROWS * LDSPAD;  // 128 x 132 floats (pong)
    float* Wl  = smem + 3 * MROWS * LDSPAD;  // 128 floats

    const int b    = blockIdx.y;
    const int kseg = blockIdx.x;
    const int tid  = threadIdx.x;
    const int wave = tid >> 5;
    const int lane = tid & 31;
    const int lm   = lane & 15;   // column / row-in-halfwave index
    const int hi   = lane >> 4;   // which half of the wave
    const int d2o  = 2 * hi;      // K-offset within a 4-wide d step (A/B layout)

    // LDS byte offsets of the two kv buffers (generic -> addrspace(3) -> offset)
    const unsigned kvbase[2] = { (unsigned)(uintptr_t)(lds_f*)KV0,
                                 (unsigned)(uintptr_t)(lds_f*)KV1 };
    // This thread's 16 (row,col) slots in the cooperative 128x128 tile copy
    // float4 index idx = tid + i*256 ; row = idx>>5 ; col = (idx&31)*4

    // ---- Stage Q_b (128x128) and weights into LDS (once per workgroup) ----
    const float4* qsrc = (const float4*)(q + (size_t)b * (MROWS * D_));
    #pragma unroll
    for (int i = 0; i < 16; ++i) {
        int idx = tid + i * 256;
        int row = idx >> 5;
        int col = (idx & 31) << 2;
        float4 v = qsrc[idx];
        *(float4*)&Qs[row * LDSPAD + col] = v;
    }
    if (tid < MROWS) Wl[tid] = weights[b * MROWS + tid];
    const int cl = context_lens[b];

    // ---- Prologue: async-issue kv chunk 0 into KV0 ----
    {
        const int blkid = block_tables[b * NBLK_ + kseg * CHUNKS];
        const float4* ksrc = (const float4*)(kv_cache + (size_t)blkid * (BLK_ * D_));
        #pragma unroll
        for (int i = 0; i < 16; ++i) {
            int idx = tid + i * 256;
            int row = idx >> 5;
            int col = (idx & 31) << 2;
            async_g2l_b128(kvbase[0] + (unsigned)((row * LDSPAD + col) << 2), ksrc + idx);
        }
    }

    for (int c = 0; c < CHUNKS; ++c) {
        // ---- Issue next chunk's async loads (double buffer), then drain chunk c ----
        if (c + 1 < CHUNKS) {
            const int blkid = block_tables[b * NBLK_ + kseg * CHUNKS + c + 1];
            const float4* ksrc = (const float4*)(kv_cache + (size_t)blkid * (BLK_ * D_));
            const unsigned nb = kvbase[(c + 1) & 1];
            #pragma unroll
            for (int i = 0; i < 16; ++i) {
                int idx = tid + i * 256;
                int row = idx >> 5;
                int col = (idx & 31) << 2;
                async_g2l_b128(nb + (unsigned)((row * LDSPAD + col) << 2), ksrc + idx);
            }
            WAIT_ASYNC(16);   // in-order: oldest 16 (= chunk c) have landed
        } else {
            WAIT_ASYNC(0);
        }
        __syncthreads();      // all waves' chunk-c data visible (Q/W ready on iter 0)

        const float* KVs = (c & 1) ? KV1 : KV0;

        // ---- 128x16x128 GEMM slab for this wave's 16 k-columns ----
        v8f acc[8] = {};  // one 16x16 f32 accumulator per m-tile

        const float* kvrow = &KVs[(wave * 16 + lm) * LDSPAD + d2o];  // B fragment base
        const float* qcol  = &Qs[lm * LDSPAD + d2o];                 // A fragment base

        #pragma unroll 4
        for (int d = 0; d < D_; d += 4) {
            v2f bb = *(const v2f*)&kvrow[d];   // B: 4(K) x 16(N) fragment
            #pragma unroll
            for (int t = 0; t < 8; ++t) {
                v2f aa = *(const v2f*)&qcol[t * 16 * LDSPAD + d];  // A: 16(M) x 4(K)
                acc[t] = __builtin_amdgcn_wmma_f32_16x16x4_f32(
                    /*neg_a=*/false, aa, /*neg_b=*/false, bb,
                    /*c_mod=*/(short)0, acc[t], /*reuse_a=*/false, /*reuse_b=*/false);
            }
        }

        // ---- relu * weight, reduce over h (rows) ----
        // acc[t][j] = score[row = t*16 + j + 8*hi, col = lm]
        float s0 = 0.f, s1 = 0.f;
        #pragma unroll
        for (int t = 0; t < 8; ++t) {
            float p = 0.f;
            #pragma unroll
            for (int j = 0; j < 8; ++j) {
                float wv = Wl[t * 16 + hi * 8 + j];     // LDS broadcast within half-wave
                p += fmaxf(acc[t][j], 0.f) * wv;
            }
            if (t < 4) s0 += p; else s1 += p;           // tiles 0..3 -> n=0, 4..7 -> n=1
        }
        // combine the two half-wave row groups (M 0..7 vs 8..15 of every tile)
        s0 += __shfl_xor(s0, 16, 32);
        s1 += __shfl_xor(s1, 16, 32);

        // ---- mask + store: lanes 0..15 write n=0, lanes 16..31 write n=1 ----
        const int col   = kseg * (CHUNKS * BLK_) + c * BLK_ + wave * 16 + lm;
        const float val = hi ? s1 : s0;
        const bool ok   = col <= (cl - N_ + hi);
        out[(size_t)(b * N_ + hi) * MAXLEN_ + col] = ok ? val : -__builtin_inff();

        __syncthreads();      // all waves done reading KVs before it is re-issued into
    }
}

extern "C" void kernel_launch(void* const* d_in, const int* in_sizes, int n_in,
                              void* d_out, int out_size, void* d_ws, size_t ws_size,
                              hipStream_t stream) {
    const float* q   = (const float*)d_in[0];
    const float* kv  = (const float*)d_in[1];
    const float* wt  = (const float*)d_in[2];
    const int*   cl  = (const int*)d_in[3];
    const int*   bt  = (const int*)d_in[4];
    // d_in[5] = max_model_len scalar (fixed at 8192 for this problem)
    float* out = (float*)d_out;

    dim3 grid(KSEG, B_);
    size_t smem = (size_t)(3 * MROWS * LDSPAD + MROWS) * sizeof(float);
    hipLaunchKernelGGL(indexer_wmma_kernel, grid, dim3(256), smem, stream,
                       q, kv, wt, cl, bt, out);
}